// AttentionMechanism_38809324487017
// MI455X (gfx1250) — compile-verified
//
#include <hip/hip_runtime.h>
#include <math.h>

// ---------------------------------------------------------------------------
// GAT-style edge attention, algebraically reduced:
//   logit[e] = <src[e], W_src@a> + <dest[e], W_dest@a> + <edge_attr[e], W_edge@a>
// then LeakyReLU(0.2) and a segment softmax over edge_index[1].
// Memory-bound: ~950 MB of edge data -> ~41 us at 23.3 TB/s.
// Dot products run on V_WMMA_F32_16X16X4_F32 (exact fp32).
// ---------------------------------------------------------------------------

typedef float v2f __attribute__((ext_vector_type(2)));
typedef float v8f __attribute__((ext_vector_type(8)));

#define IN_F 128
#define ED_F 32
#define NEG_SLOPE 0.2f
#define EPSV 1e-16f

// Order-preserving float -> uint mapping so we can use native u32 atomic max.
__device__ __forceinline__ unsigned mapOrd(float f) {
  unsigned b = __float_as_uint(f);
  return (b & 0x80000000u) ? ~b : (b | 0x80000000u);
}
__device__ __forceinline__ float unmapOrd(unsigned u) {
  return __uint_as_float((u & 0x80000000u) ? (u & 0x7FFFFFFFu) : ~u);
}
// mapOrd(-inf) == 0x007FFFFF  (identity element for the max)
#define MAPPED_NEG_INF 0x007FFFFFu

// --- Kernel 1: fold attention vector through the three weight matrices -----
// wvec[0..127]=W_src@a, wvec[128..255]=W_dest@a, wvec[256..287]=W_edge@a
__global__ void fold_weights(const float* __restrict__ Wsrc,
                             const float* __restrict__ Wdst,
                             const float* __restrict__ Wedg,
                             const float* __restrict__ avec,
                             float* __restrict__ wvec) {
  int j = blockIdx.x * blockDim.x + threadIdx.x;
  if (j >= 288) return;
  const float* row;
  if (j < 128)       row = Wsrc + (size_t)j * 128;
  else if (j < 256)  row = Wdst + (size_t)(j - 128) * 128;
  else               row = Wedg + (size_t)(j - 256) * 128;
  float s = 0.f;
#pragma unroll 8
  for (int h = 0; h < 128; ++h) s = fmaf(row[h], avec[h], s);
  wvec[j] = s;
}

// --- Kernel 2: init per-node max (mapped -inf) and sum (0) -----------------
__global__ void init_segments(unsigned* __restrict__ segmax,
                              const int* __restrict__ n_p) {
  int n = *n_p;
  float* segsum = (float*)(segmax + n);
  for (int i = blockIdx.x * blockDim.x + threadIdx.x; i < n;
       i += gridDim.x * blockDim.x) {
    segmax[i] = MAPPED_NEG_INF;
    segsum[i] = 0.f;
  }
}

// --- Kernel 3: WMMA edge logits + LeakyReLU + segment max ------------------
// One wave handles a tile of 16 edges. A-matrix (16x4 f32): lane = edge%16,
// half-wave picks K pair -> one float2 load per lane per 4-feature chunk.
// B-matrix broadcasts the folded weight chunk across all 16 columns, so
// D[m,n] = logit of edge m for every n.
__global__ void edge_scores(const float* __restrict__ src,
                            const float* __restrict__ dst,
                            const float* __restrict__ eat,
                            const int* __restrict__ eidx,   // (2,E) flat
                            const float* __restrict__ wvec, // 288 folded weights
                            float* __restrict__ s,          // E scores out
                            unsigned* __restrict__ segmax,
                            int E) {
  const int lane = threadIdx.x & 31;
  const int waveInBlock = threadIdx.x >> 5;
  const int wavesPerBlock = blockDim.x >> 5;
  const int tile = blockIdx.x * wavesPerBlock + waveInBlock;
  const int e0 = tile * 16;
  if (e0 >= E) return;  // wave-uniform: EXEC stays all-1s for the WMMAs

  const int row = lane & 15;        // edge within tile
  const int kh = (lane >> 4) << 1;  // K pair: lanes 0-15 -> {0,1}, 16-31 -> {2,3}
  int er = e0 + row;
  if (er > E - 1) er = E - 1;       // clamp (duplicate reads, guarded writes)

  const float* sR = src + (size_t)er * IN_F;
  const float* dR = dst + (size_t)er * IN_F;
  const float* aR = eat + (size_t)er * ED_F;

  v8f acc = {0.f, 0.f, 0.f, 0.f, 0.f, 0.f, 0.f, 0.f};

  for (int k = 0; k < IN_F; k += 4) {
    v2f a = *(const v2f*)(sR + k + kh);
    v2f b = *(const v2f*)(wvec + k + kh);
    acc = __builtin_amdgcn_wmma_f32_16x16x4_f32(false, a, false, b,
                                                (short)0, acc, false, false);
  }
  for (int k = 0; k < IN_F; k += 4) {
    v2f a = *(const v2f*)(dR + k + kh);
    v2f b = *(const v2f*)(wvec + 128 + k + kh);
    acc = __builtin_amdgcn_wmma_f32_16x16x4_f32(false, a, false, b,
                                                (short)0, acc, false, false);
  }
  for (int k = 0; k < ED_F; k += 4) {
    v2f a = *(const v2f*)(aR + k + kh);
    v2f b = *(const v2f*)(wvec + 256 + k + kh);
    acc = __builtin_amdgcn_wmma_f32_16x16x4_f32(false, a, false, b,
                                                (short)0, acc, false, false);
  }

  // C/D layout: VGPR j holds M=j (lanes 0-15) / M=j+8 (lanes 16-31), any N.
  // Lanes 0-7 emit edges 0-7, lanes 16-23 emit edges 8-15.
  int sub = lane & 15;
  if (sub < 8) {
    int m = sub + ((lane >> 4) << 3);
    int e = e0 + m;
    if (e < E) {
      float logit = acc[sub];
      float sc = (logit >= 0.f) ? logit : NEG_SLOPE * logit;
      s[e] = sc;
      int c = eidx[E + e];  // col = edge_index[1]
      atomicMax(&segmax[c], mapOrd(sc));
    }
  }
}

// --- Kernel 4: e = exp(s - max[col]); segment sum ---------------------------
__global__ void exp_and_sum(float* __restrict__ s,
                            const int* __restrict__ eidx,
                            unsigned* __restrict__ segmax,
                            const int* __restrict__ n_p, int E) {
  int n = *n_p;
  float* segsum = (float*)(segmax + n);
  int e = blockIdx.x * blockDim.x + threadIdx.x;
  if (e >= E) return;
  int c = eidx[E + e];
  float m = unmapOrd(segmax[c]);  // finite: col has >=1 edge
  float ev = expf(s[e] - m);
  s[e] = ev;
  atomicAdd(&segsum[c], ev);
}

// --- Kernel 5: normalize ----------------------------------------------------
__global__ void normalize_out(const float* __restrict__ s,
                              const int* __restrict__ eidx,
                              const unsigned* __restrict__ segmax,
                              const int* __restrict__ n_p,
                              float* __restrict__ out, int E) {
  int n = *n_p;
  const float* segsum = (const float*)(segmax + n);
  int e = blockIdx.x * blockDim.x + threadIdx.x;
  if (e >= E) return;
  int c = eidx[E + e];
  out[e] = s[e] / (segsum[c] + EPSV);
}

extern "C" void kernel_launch(void* const* d_in, const int* in_sizes, int n_in,
                              void* d_out, int out_size, void* d_ws, size_t ws_size,
                              hipStream_t stream) {
  const float* src  = (const float*)d_in[0];
  const float* dst  = (const float*)d_in[1];
  const float* eat  = (const float*)d_in[2];
  const int*   eidx = (const int*)d_in[3];
  const int*   n_p  = (const int*)d_in[4];   // device scalar n_nodes
  const float* Wsrc = (const float*)d_in[5];
  const float* Wdst = (const float*)d_in[6];
  const float* Wedg = (const float*)d_in[7];
  const float* avec = (const float*)d_in[8];
  float* out = (float*)d_out;

  const int E = in_sizes[0] / IN_F;

  // Workspace layout (floats): [0,512) folded weights | [512,512+E) scores |
  // then on-device: segmax[n] (u32) followed by segsum[n] (f32).
  float* wvec = (float*)d_ws;
  float* s    = wvec + 512;
  unsigned* segmax = (unsigned*)(s + E);

  fold_weights<<<1, 288, 0, stream>>>(Wsrc, Wdst, Wedg, avec, wvec);
  init_segments<<<512, 256, 0, stream>>>(segmax, n_p);

  const int wavesPerBlock = 8;  // 256 threads
  const int tiles = (E + 15) / 16;
  edge_scores<<<(tiles + wavesPerBlock - 1) / wavesPerBlock,
                32 * wavesPerBlock, 0, stream>>>(src, dst, eat, eidx, wvec, s,
                                                 segmax, E);

  const int tpb = 256;
  exp_and_sum<<<(E + tpb - 1) / tpb, tpb, 0, stream>>>(s, eidx, segmax, n_p, E);
  normalize_out<<<(E + tpb - 1) / tpb, tpb, 0, stream>>>(s, eidx, segmax, n_p,
                                                         out, E);
}